// EvNN_67121748902151
// MI455X (gfx1250) — compile-verified
//
#include <hip/hip_runtime.h>
#include <hip/hip_bf16.h>
#include <stdint.h>

#define N_NEURONS 4096
#define IN_SIZE   1024
#define K_SPIKES  8
#define N_STEPS   100
#define D_BINS    5
#define DT_F      0.01f
#define DECAY_F   0.1f    /* DT/TAU */
#define V_TH_F    1.0f

typedef float v2f __attribute__((ext_vector_type(2)));
typedef float v8f __attribute__((ext_vector_type(8)));

// ---------------- init persistent state in workspace ----------------
__global__ void k_init(float* buf, float* v, float* spk, int* counters) {
  int i = blockIdx.x * blockDim.x + threadIdx.x;
  int stride = gridDim.x * blockDim.x;
  for (int x = i; x < D_BINS * N_NEURONS; x += stride) buf[x] = 0.f;
  for (int x = i; x < N_NEURONS; x += stride) { v[x] = 0.f; spk[x] = 0.f; }
  for (int x = i; x < N_STEPS; x += stride) counters[x] = 0;
}

// ---------------- input current scatter: I_in[100][4096] ----------------
// One block owns 64 output columns; per-column bin accumulation lives in LDS,
// so there are no atomics at all (each thread owns its column).
__global__ __launch_bounds__(64) void k_input(const float* __restrict__ st,
                                              const float* __restrict__ W_in,
                                              const float* __restrict__ d_in,
                                              float* __restrict__ I_in) {
  __shared__ float sh_t[IN_SIZE * K_SPIKES];   // 32 KB
  __shared__ float sh_I[N_STEPS * 64];         // 25 KB
  int tx = threadIdx.x;
  for (int i = tx; i < IN_SIZE * K_SPIKES; i += 64) sh_t[i] = st[i];
  for (int i = tx; i < N_STEPS * 64; i += 64) sh_I[i] = 0.f;
  __syncthreads();
  int j = blockIdx.x * 64 + tx;
  for (int i = 0; i < IN_SIZE; ++i) {
    float w   = W_in[i * N_NEURONS + j];
    float dly = fmaxf(d_in[i * N_NEURONS + j], 0.f);
#pragma unroll
    for (int k = 0; k < K_SPIKES; ++k) {
      float ta = sh_t[i * K_SPIKES + k] + dly;
      int b = (int)floorf(ta / DT_F);          // match reference: floor((t-T0)/DT)
      if (b >= 0 && b < N_STEPS) sh_I[b * 64 + tx] += w;
    }
  }
  for (int r = 0; r < N_STEPS; ++r)
    I_in[r * N_NEURONS + j] = sh_I[r * 64 + tx];
}

// ---------------- pack recurrent delay bins (1..5), row-pair interleaved ----
// Pair rows (2q, 2q+1) land in adjacent bytes of u16 word [q*4096 + j] so the
// propagate kernel fetches both K-rows of a lane with one global_load_u16.
__global__ void k_dbin(const float* __restrict__ d_rec, unsigned char* __restrict__ dbp, int n) {
  int i = blockIdx.x * blockDim.x + threadIdx.x;
  if (i >= n) return;
  int p = i >> 12;                 // row (0..4095)
  int j = i & (N_NEURONS - 1);     // col
  float d = fmaxf(d_rec[i], 0.f);
  int b = (int)floorf(d / DT_F) + 1;
  b = b < 1 ? 1 : (b > D_BINS ? D_BINS : b);
  dbp[((size_t)((p >> 1) * N_NEURONS + j) << 1) + (p & 1)] = (unsigned char)b;
}

// ---------------- per-step neuron update ----------------
__global__ void k_step(const float* __restrict__ I_in, float* __restrict__ buf,
                       float* __restrict__ v, float* __restrict__ spk,
                       int* __restrict__ counters, float* __restrict__ out, int t) {
  int j = blockIdx.x * blockDim.x + threadIdx.x;   // 4096 threads
  int slot = t % D_BINS;
  float I  = I_in[t * N_NEURONS + j] + buf[slot * N_NEURONS + j];
  float vv = v[j];
  float vn = vv + DECAY_F * (-vv) + I;
  float sp = (vn >= V_TH_F) ? 1.0f : 0.0f;
  float vo = (sp > 0.f) ? 0.f : vn;
  out[t * N_NEURONS + j] = vo;
  v[j]   = vo;
  spk[j] = sp;
  buf[slot * N_NEURONS + j] = 0.f;                 // reference zeroes slot before scatter
  unsigned m = __builtin_amdgcn_ballot_w32(sp > 0.f);
  if ((threadIdx.x & 31) == 0) atomicAdd(&counters[t], (int)__popc(m));
}

// ---------------- per-step recurrent propagate: 5 masked GEMVs via f32 WMMA ----
// acc_d[j] = sum_p spike[p] * W_rec[p][j] * [dbin[p][j] == d],  d = 1..5
// (accumulators indexed by DELAY; ring-slot remap (t5+d)%5 done once at writeout)
// A (16x4 f32): spike values replicated across M rows -> all D rows identical,
// row 0 extracted from acc VGPR0 of lanes 0..15.  One wave owns one 16-col tile.
__global__ __launch_bounds__(256) void k_prop(const float* __restrict__ Wrec,
                                              const unsigned short* __restrict__ dbp,
                                              const float* __restrict__ spk,
                                              const int* __restrict__ counters,
                                              float* __restrict__ buf, int t) {
  if (counters[t] == 0) return;                    // uniform: silent step, ring buffer unchanged
  __shared__ float s_spk[N_NEURONS];               // 16 KB
  int tx = threadIdx.x;

  // CDNA5 async global->LDS staging of the spike vector (ASYNCcnt path)
  {
    unsigned lds = (unsigned)(uintptr_t)(&s_spk[0]) + (unsigned)(tx * 16);
    const float* g = spk + tx * 4;
#pragma unroll
    for (int it = 0; it < 4; ++it) {
      asm volatile("global_load_async_to_lds_b128 %0, %1, off"
                   :: "v"(lds), "v"(g) : "memory");
      lds += 256 * 16;
      g   += 256 * 4;
    }
    asm volatile("s_wait_asynccnt 0" ::: "memory");
  }
  __syncthreads();

  int wave = tx >> 5;
  int lane = tx & 31;
  int tile = blockIdx.x * 8 + wave;                // 32 blocks * 8 waves = 256 tiles
  int n    = lane & 15;
  int hi   = lane >> 4;
  int j    = tile * 16 + n;
  int t5   = t % D_BINS;

  v8f a1 = {}, a2 = {}, a3 = {}, a4 = {}, a5 = {};  // per-delay accumulators

  for (int p0 = 0; p0 < N_NEURONS; p0 += 4) {
    // A layout 16x4 f32: VGPR0 -> K = 2*hi, VGPR1 -> K = 2*hi+1 (rows replicated)
    v2f a;
    a.x = s_spk[p0 + 2 * hi];
    a.y = s_spk[p0 + 2 * hi + 1];
    // event-driven skip: whole 4-row K-tile silent -> no contribution
    if (__builtin_amdgcn_ballot_w32(a.x != 0.f || a.y != 0.f) == 0u) continue;

    int r0 = (p0 + 2 * hi) * N_NEURONS + j;
    __builtin_prefetch(&Wrec[r0 + 8 * N_NEURONS], 0, 3);  // keep near caches (WGP scope)
    float w0 = Wrec[r0];
    float w1 = Wrec[r0 + N_NEURONS];
    unsigned dd = dbp[((p0 >> 1) + hi) * N_NEURONS + j];  // packed row-pair delay bins
    int d0 = (int)(dd & 0xffu);
    int d1 = (int)(dd >> 8);

    v2f b;
    b.x = (d0 == 1) ? w0 : 0.f;  b.y = (d1 == 1) ? w1 : 0.f;
    a1 = __builtin_amdgcn_wmma_f32_16x16x4_f32(false, a, false, b, (short)0, a1, false, false);
    b.x = (d0 == 2) ? w0 : 0.f;  b.y = (d1 == 2) ? w1 : 0.f;
    a2 = __builtin_amdgcn_wmma_f32_16x16x4_f32(false, a, false, b, (short)0, a2, false, false);
    b.x = (d0 == 3) ? w0 : 0.f;  b.y = (d1 == 3) ? w1 : 0.f;
    a3 = __builtin_amdgcn_wmma_f32_16x16x4_f32(false, a, false, b, (short)0, a3, false, false);
    b.x = (d0 == 4) ? w0 : 0.f;  b.y = (d1 == 4) ? w1 : 0.f;
    a4 = __builtin_amdgcn_wmma_f32_16x16x4_f32(false, a, false, b, (short)0, a4, false, false);
    b.x = (d0 == 5) ? w0 : 0.f;  b.y = (d1 == 5) ? w1 : 0.f;
    a5 = __builtin_amdgcn_wmma_f32_16x16x4_f32(false, a, false, b, (short)0, a5, false, false);
  }

  // D row 0 lives in accumulator VGPR0 of lanes 0..15; tile columns are exclusive
  // to this wave, so plain read-modify-write on the ring buffer is race-free.
  if (lane < 16) {
    int jj = tile * 16 + lane;
    int s1 = t5 + 1; s1 -= (s1 >= D_BINS) ? D_BINS : 0;
    int s2 = t5 + 2; s2 -= (s2 >= D_BINS) ? D_BINS : 0;
    int s3 = t5 + 3; s3 -= (s3 >= D_BINS) ? D_BINS : 0;
    int s4 = t5 + 4; s4 -= (s4 >= D_BINS) ? D_BINS : 0;
    int s5 = t5;                                      // (t5+5)%5
    buf[s1 * N_NEURONS + jj] += a1[0];
    buf[s2 * N_NEURONS + jj] += a2[0];
    buf[s3 * N_NEURONS + jj] += a3[0];
    buf[s4 * N_NEURONS + jj] += a4[0];
    buf[s5 * N_NEURONS + jj] += a5[0];
  }
}

extern "C" void kernel_launch(void* const* d_in, const int* in_sizes, int n_in,
                              void* d_out, int out_size, void* d_ws, size_t ws_size,
                              hipStream_t stream) {
  (void)in_sizes; (void)n_in; (void)out_size; (void)ws_size;
  const float* st = (const float*)d_in[0];   // (1024, 8)
  const float* W  = (const float*)d_in[1];   // (5120, 4096)
  const float* dl = (const float*)d_in[2];   // (5120, 4096)
  float* out = (float*)d_out;                // (100, 4096)

  char* ws = (char*)d_ws;
  float* I_in = (float*)(ws + 0);            // 100*4096 f32   = 1638400 B
  float* buf  = (float*)(ws + 1638400);      // 5*4096 f32     =   81920 B
  float* v    = (float*)(ws + 1720320);      // 4096 f32
  float* spk  = (float*)(ws + 1736704);      // 4096 f32
  int*   cnt  = (int*)  (ws + 1753088);      // 100 i32 (padded to 512)
  unsigned char* dbp = (unsigned char*)(ws + 1753600);  // 4096*4096 u8 (pair-packed)

  const float* W_in  = W  + (size_t)N_NEURONS * N_NEURONS;
  const float* d_inp = dl + (size_t)N_NEURONS * N_NEURONS;

  k_init<<<32, 256, 0, stream>>>(buf, v, spk, cnt);
  k_input<<<N_NEURONS / 64, 64, 0, stream>>>(st, W_in, d_inp, I_in);
  k_dbin<<<(N_NEURONS * N_NEURONS + 255) / 256, 256, 0, stream>>>(dl, dbp, N_NEURONS * N_NEURONS);

  for (int t = 0; t < N_STEPS; ++t) {
    k_step<<<N_NEURONS / 256, 256, 0, stream>>>(I_in, buf, v, spk, cnt, out, t);
    k_prop<<<32, 256, 0, stream>>>(W, (const unsigned short*)dbp, spk, cnt, buf, t);
  }
}